// RecNN_70574902608433
// MI455X (gfx1250) — compile-verified
//
#include <hip/hip_runtime.h>
#include <stdint.h>

// ---------------- problem geometry ----------------
constexpr int W_NODES = 262144;           // nodes per level
constexpr int L_LEV   = 32;               // levels
constexpr int TPB     = 256;              // 8 wave32 per block
constexpr int BLOCKS  = 128;
constexpr int NPT     = 4;                // node-quarters per thread per tile
constexpr int NODES_PER_ITER_BLOCK = TPB * NPT;                 // 1024
constexpr int NODES_PER_BLOCK      = W_NODES / BLOCKS;          // 2048
constexpr int NITER   = NODES_PER_BLOCK / NODES_PER_ITER_BLOCK; // 2 (double buffered)
static_assert(NITER == 2, "pipeline below is hand-unrolled for 2 tiles");
constexpr int WAVE_NODES = 32 * NPT;      // 128 nodes per wave-tile
constexpr int IDX_B   = WAVE_NODES * 8;   // 1024 B child indices per wave-tile
constexpr int TYP_B   = WAVE_NODES * 4;   // 512 B node types per wave-tile
constexpr int BUF_B   = IDX_B + TYP_B;    // 1536 B per buffer
constexpr int NWAVES  = TPB / 32;         // 8

typedef __attribute__((ext_vector_type(2))) float v2f;
typedef __attribute__((ext_vector_type(8))) float v8f;

// gfx1250 async global->LDS copy (ASYNCcnt-tracked, GVS addressing:
// 64-bit SGPR base + 32-bit per-lane VGPR byte offset; dst VGPR = LDS byte addr)
__device__ __forceinline__ void async_ld_b128(uint32_t lds_addr, uint32_t voff,
                                              const void* sbase) {
    asm volatile("global_load_async_to_lds_b128 %0, %1, %2"
                 :: "v"(lds_addr), "v"(voff), "s"(sbase)
                 : "memory");
}
__device__ __forceinline__ void wait_async_le3() {
    asm volatile("s_wait_asynccnt 0x3" ::: "memory");
}
__device__ __forceinline__ void wait_async_le0() {
    asm volatile("s_wait_asynccnt 0x0" ::: "memory");
}

// ---------------- level 0: out = values * w_term ----------------
__global__ __launch_bounds__(TPB) void level0_kernel(const float* __restrict__ values,
                                                     const float* __restrict__ w_term,
                                                     float* __restrict__ outs) {
    const float wt = w_term[0];
    const int t = blockIdx.x * (TPB * 8) + threadIdx.x * 8;  // 128 blocks cover W
    float4 a = *(const float4*)(values + t);
    float4 b = *(const float4*)(values + t + 4);
    a.x *= wt; a.y *= wt; a.z *= wt; a.w *= wt;
    b.x *= wt; b.y *= wt; b.z *= wt; b.w *= wt;
    *(float4*)(outs + t)     = a;
    *(float4*)(outs + t + 4) = b;
}

// per-tile register state: gathered children + types, kept live across tiles.
// node handled by (quarter q, lane) = nbase + q*32 + lane
struct TileVals {
    float x0[NPT], x1[NPT];
    int   typ[NPT];
    int   nbase;
};

// ---------------- levels 1..31 ----------------
__global__ __launch_bounds__(TPB) void level_kernel(
    const int*   __restrict__ idx_lvl,   // child_idx slab for this level  [W*2]
    const int*   __restrict__ typ_lvl,   // node_types slab for this level [W]
    float*       __restrict__ outs,      // [L*W] node outputs (32 MB, L2-resident)
    const float* __restrict__ w_plus,
    const float* __restrict__ w_minus,
    const float* __restrict__ w_final,
    const float* __restrict__ b_final,
    float*       __restrict__ d_final,
    int lvl)
{
    __shared__ __align__(16) char smem[NWAVES * 2 * BUF_B];  // 24 KB

    const int tid  = threadIdx.x;
    const int wave = tid >> 5;
    const int lane = tid & 31;
    const int block_base = blockIdx.x * NODES_PER_BLOCK;

    // uniform weight loads -> s_load
    const float wp0 = w_plus[0],  wp1 = w_plus[1];
    const float wm0 = w_minus[0], wm1 = w_minus[1];
    float* out_lvl = outs + (size_t)lvl * W_NODES;

    // Constant A matrix for v_wmma_f32_16x16x4_f32:
    //   row M=0 = [1,1,0,0], row M=8 = [0,0,1,1], all other rows zero.
    // A layout (16x4 f32): VGPR0 = {K0 | K2}, VGPR1 = {K1 | K3}, M = lane%16.
    //   A[0][0],A[0][1] -> lane 0 of both VGPRs; A[8][2],A[8][3] -> lane 24.
    // With B VGPR0/1 = per-lane {b0,b1} (B layout {K0|K2},{K1|K3}, node = lane),
    //   D.vgpr0[lane] = b0(lane)+b1(lane): 32 node results, already per-lane.
    const float aval = (lane == 0 || lane == 24) ? 1.0f : 0.0f;
    v2f a2; a2.x = aval; a2.y = aval;
    const v8f czero = {};

    // LDS byte address of this wave's staging area (flat-addr low 32 bits = LDS offset)
    const uint32_t lds_wave = (uint32_t)(uintptr_t)(&smem[wave * 2 * BUF_B]);

    auto issue = [&](int it) {
        const int buf = it & 1;
        const uint32_t lbase = lds_wave + buf * BUF_B;
        const int wavebase = block_base + it * NODES_PER_ITER_BLOCK + wave * WAVE_NODES;
        // child indices: 1024 B per wave-tile -> 2 x b128 per lane
        const uint32_t g_idx = (uint32_t)(wavebase * 8) + (uint32_t)(lane * 16);
        async_ld_b128(lbase + lane * 16,       g_idx,       idx_lvl);
        async_ld_b128(lbase + 512 + lane * 16, g_idx + 512, idx_lvl);
        // node types: 512 B per wave-tile -> 1 x b128 per lane
        const uint32_t g_typ = (uint32_t)(wavebase * 4) + (uint32_t)(lane * 16);
        async_ld_b128(lbase + IDX_B + lane * 16, g_typ, typ_lvl);
    };

    // read staged idx/types from LDS and put this tile's 8 gathers in flight;
    // results are consumed later so loads stay pending across the other tile
    auto load_tile = [&](int it) -> TileVals {
        const int buf = it & 1;
        const char* base = &smem[wave * 2 * BUF_B + buf * BUF_B];
        const int2* idx2 = (const int2*)base;            // ds_load_b64, stride 8B: no bank conflicts
        const int*  typs = (const int*)(base + IDX_B);   // ds_load_b32, stride 4B
        TileVals tv;
        tv.nbase = block_base + it * NODES_PER_ITER_BLOCK + wave * WAVE_NODES;
#pragma unroll
        for (int q = 0; q < NPT; ++q) {
            const int2 ii = idx2[q * 32 + lane];
            tv.typ[q] = typs[q * 32 + lane];
            tv.x0[q] = outs[ii.x];                       // L2-resident gathers
            tv.x1[q] = outs[ii.y];
        }
        return tv;
    };

    auto finish = [&](const TileVals& tv) {
#pragma unroll
        for (int q = 0; q < NPT; ++q) {
            const bool p = (tv.typ[q] == 1);
            // per-lane selected products = B matrix rows for this quarter's 32 nodes
            v2f b;
            b.x = (p ? wp0 : wm0) * tv.x0[q];
            b.y = (p ? wp1 : wm1) * tv.x1[q];
            // matrix pipe performs the 2-term reduction; result lands per-lane in d[0]
            v8f d = __builtin_amdgcn_wmma_f32_16x16x4_f32(
                false, a2, false, b, (short)0, czero, false, false);
            const float o = d[0];
            const int node = tv.nbase + q * 32 + lane;
            out_lvl[node] = o;                           // coalesced b32 store

            // fold the final Linear(1,1,bias=True) into the owner of node N-1
            if (lvl == L_LEV - 1 && node == W_NODES - 1) {
                d_final[0] = fmaf(o, w_final[0], b_final[0]);
            }
        }
    };

    // two-tile software pipeline:
    //   6 async b128 copies in flight, then 16 gathers/thread in flight
    issue(0);
    issue(1);
    wait_async_le3();               // in-order completion: tile 0 staged
    TileVals t0 = load_tile(0);     // 8 gathers issued
    wait_async_le0();               // tile 1 staged
    TileVals t1 = load_tile(1);     // 8 more gathers issued (16 total in flight)
    finish(t0);                     // compiler waits loadcnt<=8 here
    finish(t1);                     // then loadcnt<=0
}

// ---------------- host launch ----------------
extern "C" void kernel_launch(void* const* d_in, const int* in_sizes, int n_in,
                              void* d_out, int out_size, void* d_ws, size_t ws_size,
                              hipStream_t stream) {
    const float* values     = (const float*)d_in[0];
    const int*   child_idx  = (const int*)  d_in[1];  // [(L-1), W, 2]
    const int*   node_types = (const int*)  d_in[2];  // [(L-1), W]
    const float* w_term     = (const float*)d_in[3];
    const float* w_plus     = (const float*)d_in[4];
    const float* w_minus    = (const float*)d_in[5];
    const float* w_final    = (const float*)d_in[6];
    const float* b_final    = (const float*)d_in[7];

    float* outs = (float*)d_ws;          // L*W floats = 32 MB scratch
    float* out  = (float*)d_out;         // single f32

    level0_kernel<<<BLOCKS, TPB, 0, stream>>>(values, w_term, outs);

    for (int lvl = 1; lvl < L_LEV; ++lvl) {
        level_kernel<<<BLOCKS, TPB, 0, stream>>>(
            child_idx + (size_t)(lvl - 1) * W_NODES * 2,
            node_types + (size_t)(lvl - 1) * W_NODES,
            outs, w_plus, w_minus, w_final, b_final, out, lvl);
    }
}